// GAT2_18863496364167
// MI455X (gfx1250) — compile-verified
//
#include <hip/hip_runtime.h>

#define N_NODES   50000
#define N_EDGES   800000
#define DIM_IN    128
#define HC        256      // HEADS * per-head channels (both layers)
#define CH        64       // per-head channels (HID == EMB == 64)
#define HEADS     4
#define OUT_DIM   2
#define NEG_SLOPE 0.2f

typedef __attribute__((ext_vector_type(2))) float v2f;
typedef __attribute__((ext_vector_type(8))) float v8f;

// ---------------------------------------------------------------------------
// Monotone float <-> uint encoding for atomicMax-based segment max
// ---------------------------------------------------------------------------
__device__ __forceinline__ unsigned fenc(float f) {
  unsigned u = __float_as_uint(f);
  return (u & 0x80000000u) ? ~u : (u | 0x80000000u);
}
__device__ __forceinline__ float fdec(unsigned k) {
  return (k & 0x80000000u) ? __uint_as_float(k ^ 0x80000000u)
                           : __uint_as_float(~k);
}

// ---------------------------------------------------------------------------
// WMMA fp32 GEMM:  C[M,Nout] = A[M,K] @ W[K,Nout] + bias[Nout]
// One wave per 16x16 output tile, K-loop in steps of 4 via
// V_WMMA_F32_16X16X4_F32. Requires M%16==0, K%4==0, Nout%16==0 (holds for all
// call sites), so EXEC is all-ones everywhere (WMMA requirement).
// ---------------------------------------------------------------------------
__global__ __launch_bounds__(256) void gemm_wmma_f32(
    const float* __restrict__ A, const float* __restrict__ W,
    const float* __restrict__ bias, float* __restrict__ C,
    int K, int Nout)
{
  const int wave     = (blockIdx.x << 3) + (threadIdx.x >> 5);
  const int ntilesN  = Nout >> 4;
  const int tn       = wave % ntilesN;
  const int tm       = wave / ntilesN;
  const int lane     = threadIdx.x & 31;
  const int lm       = lane & 15;   // M index for A-frag, N index for B/C-frag
  const int hi       = lane >> 4;   // K half-select

  const int  row = (tm << 4) + lm;          // A row this lane loads
  const int  col = (tn << 4) + lm;          // C/B column this lane owns
  const float bv = bias[col];
  v8f acc = {bv, bv, bv, bv, bv, bv, bv, bv};

  const float* arow = A + (size_t)row * K + (hi << 1);          // K=2*hi, 2*hi+1
  const float* wcol = W + (size_t)(hi << 1) * Nout + col;
  const size_t wstep = (size_t)Nout;

  for (int kk = 0; kk < K; kk += 4) {
    v2f a = *reinterpret_cast<const v2f*>(arow + kk);           // contiguous pair
    v2f b;
    b.x = wcol[(size_t)kk * wstep];
    b.y = wcol[(size_t)kk * wstep + wstep];
    acc = __builtin_amdgcn_wmma_f32_16x16x4_f32(
        /*neg_a=*/false, a, /*neg_b=*/false, b,
        /*c_mod=*/(short)0, acc, /*reuse_a=*/false, /*reuse_b=*/false);
  }

  // C/D layout: VGPR j holds M = j (lanes 0-15) / M = j+8 (lanes 16-31)
  float* crow = C + (size_t)((tm << 4) + (hi << 3)) * Nout + col;
  #pragma unroll
  for (int j = 0; j < 8; ++j)
    crow[(size_t)j * Nout] = acc[j];
}

// ---------------------------------------------------------------------------
// Per-(edge,head) attention logit + running segment max (encoded atomicMax)
// ---------------------------------------------------------------------------
__global__ __launch_bounds__(256) void edge_logit_kernel(
    const int* __restrict__ src, const int* __restrict__ dst,
    const float* __restrict__ xl, const float* __restrict__ xr,
    const float* __restrict__ att,
    float* __restrict__ logit, unsigned* __restrict__ menc)
{
  int t = blockIdx.x * blockDim.x + threadIdx.x;
  if (t >= N_EDGES * HEADS) return;
  const int e = t >> 2, h = t & 3;
  const int s = src[e], d = dst[e];

  const float4* l = (const float4*)(xl + (size_t)s * HC + h * CH);
  const float4* r = (const float4*)(xr + (size_t)d * HC + h * CH);
  const float4* a = (const float4*)(att + h * CH);
  float acc = 0.f;
  #pragma unroll 4
  for (int c = 0; c < CH / 4; ++c) {
    float4 lv = l[c], rv = r[c], av = a[c];
    float s0 = lv.x + rv.x, s1 = lv.y + rv.y;
    float s2 = lv.z + rv.z, s3 = lv.w + rv.w;
    s0 = fmaxf(s0, 0.f) + NEG_SLOPE * fminf(s0, 0.f);
    s1 = fmaxf(s1, 0.f) + NEG_SLOPE * fminf(s1, 0.f);
    s2 = fmaxf(s2, 0.f) + NEG_SLOPE * fminf(s2, 0.f);
    s3 = fmaxf(s3, 0.f) + NEG_SLOPE * fminf(s3, 0.f);
    acc += s0 * av.x + s1 * av.y + s2 * av.z + s3 * av.w;
  }
  logit[t] = acc;
  atomicMax(&menc[d * HEADS + h], fenc(acc));
}

// ---------------------------------------------------------------------------
// p = exp(logit - segmax[dst]); accumulate softmax denominators
// ---------------------------------------------------------------------------
__global__ __launch_bounds__(256) void edge_exp_kernel(
    const int* __restrict__ dst, float* __restrict__ logit,
    const unsigned* __restrict__ menc, float* __restrict__ denom)
{
  int t = blockIdx.x * blockDim.x + threadIdx.x;
  if (t >= N_EDGES * HEADS) return;
  const int e = t >> 2, h = t & 3;
  const int d = dst[e];
  float p = expf(logit[t] - fdec(menc[d * HEADS + h]));
  logit[t] = p;                       // reuse buffer: now holds p
  atomicAdd(&denom[d * HEADS + h], p);
}

// ---------------------------------------------------------------------------
// agg[dst] += (p/denom[dst]) * xl[src]   (scatter with fp32 atomics into L2)
// ---------------------------------------------------------------------------
__global__ __launch_bounds__(256) void edge_agg_kernel(
    const int* __restrict__ src, const int* __restrict__ dst,
    const float* __restrict__ xl, const float* __restrict__ p,
    const float* __restrict__ denom, float* __restrict__ agg)
{
  int t = blockIdx.x * blockDim.x + threadIdx.x;
  if (t >= N_EDGES * HEADS) return;
  const int e = t >> 2, h = t & 3;
  const int s = src[e], d = dst[e];
  const float alpha = p[t] / denom[d * HEADS + h];
  const float4* xs = (const float4*)(xl + (size_t)s * HC + h * CH);
  float* o = agg + (size_t)d * HC + h * CH;
  #pragma unroll 4
  for (int c = 0; c < CH / 4; ++c) {
    float4 v = xs[c];
    atomicAdd(o + 4 * c + 0, alpha * v.x);
    atomicAdd(o + 4 * c + 1, alpha * v.y);
    atomicAdd(o + 4 * c + 2, alpha * v.z);
    atomicAdd(o + 4 * c + 3, alpha * v.w);
  }
}

// ---------------------------------------------------------------------------
// h = relu(agg + bias1)   (in place, float4)
// ---------------------------------------------------------------------------
__global__ __launch_bounds__(256) void bias_relu_kernel(
    float* __restrict__ h, const float* __restrict__ bias)
{
  int t = blockIdx.x * blockDim.x + threadIdx.x;
  if (t >= N_NODES * HC / 4) return;
  float4 v = ((float4*)h)[t];
  float4 b = ((const float4*)bias)[t & (HC / 4 - 1)];
  v.x = fmaxf(v.x + b.x, 0.f);
  v.y = fmaxf(v.y + b.y, 0.f);
  v.z = fmaxf(v.z + b.z, 0.f);
  v.w = fmaxf(v.w + b.w, 0.f);
  ((float4*)h)[t] = v;
}

// ---------------------------------------------------------------------------
// hm[n,c] = mean over heads of agg2[n,h,c] + bias2[c]
// ---------------------------------------------------------------------------
__global__ __launch_bounds__(256) void mean_bias_kernel(
    const float* __restrict__ agg2, const float* __restrict__ bias2,
    float* __restrict__ hm)
{
  int t = blockIdx.x * blockDim.x + threadIdx.x;
  if (t >= N_NODES * CH) return;
  const int n = t / CH, c = t - n * CH;
  const float* r = agg2 + (size_t)n * HC;
  hm[t] = 0.25f * (r[c] + r[CH + c] + r[2 * CH + c] + r[3 * CH + c]) + bias2[c];
}

// ---------------------------------------------------------------------------
// out[n,:] = hm[n,:] @ Wout + bout     (64x2 — too narrow for WMMA)
// ---------------------------------------------------------------------------
__global__ __launch_bounds__(256) void out_kernel(
    const float* __restrict__ hm, const float* __restrict__ Wout,
    const float* __restrict__ bout, float* __restrict__ out)
{
  int n = blockIdx.x * blockDim.x + threadIdx.x;
  if (n >= N_NODES) return;
  const float* r = hm + (size_t)n * CH;
  float a0 = bout[0], a1 = bout[1];
  #pragma unroll 8
  for (int c = 0; c < CH; ++c) {
    float v = r[c];
    a0 += v * Wout[2 * c + 0];
    a1 += v * Wout[2 * c + 1];
  }
  out[2 * n + 0] = a0;
  out[2 * n + 1] = a1;
}

// ---------------------------------------------------------------------------
extern "C" void kernel_launch(void* const* d_in, const int* in_sizes, int n_in,
                              void* d_out, int out_size, void* d_ws, size_t ws_size,
                              hipStream_t stream) {
  (void)in_sizes; (void)n_in; (void)out_size; (void)ws_size;

  const float* x     = (const float*)d_in[0];
  const int*   ei    = (const int*)d_in[1];
  const int*   src   = ei;
  const int*   dst   = ei + N_EDGES;
  const float* Wl1   = (const float*)d_in[2];
  const float* bl1   = (const float*)d_in[3];
  const float* Wr1   = (const float*)d_in[4];
  const float* br1   = (const float*)d_in[5];
  const float* att1  = (const float*)d_in[6];
  const float* bias1 = (const float*)d_in[7];
  const float* Wl2   = (const float*)d_in[8];
  const float* bl2   = (const float*)d_in[9];
  const float* Wr2   = (const float*)d_in[10];
  const float* br2   = (const float*)d_in[11];
  const float* att2  = (const float*)d_in[12];
  const float* bias2 = (const float*)d_in[13];
  const float* Wout  = (const float*)d_in[14];
  const float* bout  = (const float*)d_in[15];
  float* out = (float*)d_out;

  // Workspace carve-up (~168 MB): three N x 256 feature buffers stay L2-resident.
  const size_t FEAT = (size_t)N_NODES * HC * sizeof(float);   // 51.2 MB
  const size_t ELOG = (size_t)N_EDGES * HEADS * sizeof(float);// 12.8 MB
  const size_t SEG  = (size_t)N_NODES * HEADS * sizeof(float);// 0.8 MB
  char* ws = (char*)d_ws;
  float*    A  = (float*)(ws);                 // xl (layer 1 then layer 2)
  float*    B  = (float*)(ws + FEAT);          // xr ; later hm
  float*    Cb = (float*)(ws + 2 * FEAT);      // aggregation / h1
  float*    L  = (float*)(ws + 3 * FEAT);      // logits -> p
  unsigned* M  = (unsigned*)(ws + 3 * FEAT + ELOG);          // encoded seg-max
  float*    D  = (float*)(ws + 3 * FEAT + ELOG + SEG);       // denominators

  const int gemmBlocks = (N_NODES / 16) * (HC / 16) / 8;     // 6250, exact
  const int eThreads   = N_EDGES * HEADS;
  const int eBlocks    = (eThreads + 255) / 256;             // 12500

  // ---- Layer 1 ----
  gemm_wmma_f32<<<gemmBlocks, 256, 0, stream>>>(x, Wl1, bl1, A, DIM_IN, HC);
  gemm_wmma_f32<<<gemmBlocks, 256, 0, stream>>>(x, Wr1, br1, B, DIM_IN, HC);
  hipMemsetAsync(M, 0, SEG, stream);
  hipMemsetAsync(D, 0, SEG, stream);
  hipMemsetAsync(Cb, 0, FEAT, stream);
  edge_logit_kernel<<<eBlocks, 256, 0, stream>>>(src, dst, A, B, att1, L, M);
  edge_exp_kernel<<<eBlocks, 256, 0, stream>>>(dst, L, M, D);
  edge_agg_kernel<<<eBlocks, 256, 0, stream>>>(src, dst, A, L, D, Cb);
  bias_relu_kernel<<<(N_NODES * HC / 4 + 255) / 256, 256, 0, stream>>>(Cb, bias1);

  // ---- Layer 2 (h1 lives in Cb; A/B are free to reuse) ----
  gemm_wmma_f32<<<gemmBlocks, 256, 0, stream>>>(Cb, Wl2, bl2, A, HC, HC);
  gemm_wmma_f32<<<gemmBlocks, 256, 0, stream>>>(Cb, Wr2, br2, B, HC, HC);
  hipMemsetAsync(M, 0, SEG, stream);
  hipMemsetAsync(D, 0, SEG, stream);
  edge_logit_kernel<<<eBlocks, 256, 0, stream>>>(src, dst, A, B, att2, L, M);
  edge_exp_kernel<<<eBlocks, 256, 0, stream>>>(dst, L, M, D);
  hipMemsetAsync(Cb, 0, FEAT, stream);   // after GEMMs consumed h1
  edge_agg_kernel<<<eBlocks, 256, 0, stream>>>(src, dst, A, L, D, Cb);

  // ---- Head mean + bias2, then 64x2 projection ----
  mean_bias_kernel<<<(N_NODES * CH + 255) / 256, 256, 0, stream>>>(Cb, bias2, B);
  out_kernel<<<(N_NODES + 255) / 256, 256, 0, stream>>>(B, Wout, bout, out);
}